// RTPSphericalHarmonicsPyT_11347303596044
// MI455X (gfx1250) — compile-verified
//
#include <hip/hip_runtime.h>
#include <math.h>

// ---------------------------------------------------------------------------
// Spherical harmonics (l<=20, K=441) for two 3-vectors per point, then
// out[n,r] = Psi1[n, rand_i[r]] * Psi2[n, rand_j[r]],  r < 256.
//
// Design (MI455X / gfx1250, wave32):
//  * per-thread point; per-thread LDS slice of 545 words (odd stride ->
//    conflict-free across 32 lanes on 64 banks):
//      [0..230]   P(l,m) vector 1      (m-major chains)
//      [231..271] cos(m a),sin(m a) v1 (cos at +m, sin at +20+m)
//      [272..502] P(l,m) vector 2
//      [503..543] trig vector 2
//  * init kernel folds BOTH norm constants (fp64) and all four LDS indices
//    per rank-slot into d_ws: int4 idx[256] + float coef[256].
//  * coordinates staged via CDNA5 async global->LDS (coalesced b32 copies),
//    s_wait_asynccnt + workgroup barrier.
//  * Legendre computed row-wise (l outer), both vectors interleaved, fully
//    unrolled => Pa/Pb[2][21] in VGPRs, ILP ~ 2*(l+1) per step.
//  * STATIC shared (141,056 B) -> group segment size verified at compile time,
//    2 workgroups / WGP (282 KB of 320 KB), 1 wave per SIMD32, ILP-covered.
// ---------------------------------------------------------------------------

#define LMAX   20
#define NP     231           // # P(l,m) entries with m>=0
#define VOFF   272           // per-vector region: 231 P + 41 trig
#define SLICE  545           // 2*272 = 544, padded to odd word stride
#define TPB    64
#define RANKN  256
#define STAGE_OFF (TPB*SLICE)          // staging region (words) after slices
#define LDS_WORDS (TPB*SLICE + TPB*6)  // 34880 + 384 = 35264 words = 141056 B

__host__ __device__ constexpr int col_off(int m) { return m*(43 - m)/2; } // sum_{m'<m}(21-m')

// ------------------------- metadata init kernel ----------------------------
__global__ void sh_meta(const int* __restrict__ ri, const int* __restrict__ rj,
                        int4* __restrict__ midx, float* __restrict__ mcoef)
{
    int r = threadIdx.x;
    if (r >= RANKN) return;
    int kk2[2] = { ri[r], rj[r] };
    int pi[2], ti[2];
    double c[2];
    for (int s = 0; s < 2; ++s) {
        int kk = kk2[s];
        int l = 0;
        while ((l+1)*(l+1) <= kk) ++l;          // l = floor(sqrt(kk))
        int m  = kk - l*(l+1);                  // signed m
        int am = (m < 0) ? -m : m;
        pi[s] = col_off(am) + (l - am);
        ti[s] = (m == 0) ? 0 : ((m > 0) ? am : 20 + am);
        double ratio = 1.0;                     // (l-am)! / (l+am)!
        for (int t = l-am+1; t <= l+am; ++t) ratio /= (double)t;
        double cc = sqrt((double)(2*l+1) / (4.0*M_PI) * ratio);
        if (am) cc *= 1.4142135623730951;       // sqrt(2) for m != 0
        c[s] = cc;
    }
    midx[r]  = make_int4(pi[0], NP + ti[0], VOFF + pi[1], VOFF + NP + ti[1]);
    mcoef[r] = (float)(c[0]*c[1]);
}

// ------------------------------ main kernel --------------------------------
__global__ __launch_bounds__(TPB) void sh_main(
    const float* __restrict__ coords,
    const int4*  __restrict__ midx,
    const float* __restrict__ mcoef,
    float*       __restrict__ out,
    int npts)
{
    __shared__ float smem[LDS_WORDS];          // 141,056 B, compile-checked
    const int    tid = threadIdx.x;
    const size_t p0  = (size_t)blockIdx.x * TPB;

    // ---- CDNA5 async global->LDS staging of this block's 64x6 coords ------
    {
        const float* gsrc = coords + p0*6;
        long remain = (long)npts*6 - (long)p0*6;
        if (remain > TPB*6) remain = TPB*6;
        float* stage = smem + STAGE_OFF;
        #pragma unroll
        for (int kk = 0; kk < 6; ++kk) {
            int w = tid + kk*TPB;                  // coalesced word index
            if (w < (int)remain) {
                unsigned ldsb = (unsigned)(size_t)(&stage[w]); // LDS byte offset
                asm volatile("global_load_async_to_lds_b32 %0, %1, %2 offset:0"
                             :: "v"(ldsb), "v"(w*4), "s"(gsrc) : "memory");
            }
        }
        asm volatile("s_wait_asynccnt 0" ::: "memory");
    }
    __syncthreads();

    const size_t n      = p0 + tid;
    const bool   active = (n < (size_t)npts);

    float xx[2], yy[2], zz[2];
    {
        const float* my = smem + STAGE_OFF + tid*6;
        xx[0]=my[0]; yy[0]=my[1]; zz[0]=my[2];
        xx[1]=my[3]; yy[1]=my[4]; zz[1]=my[5];
    }

    // ---- direction params, no transcendentals:
    //   ct = cos(inc) = z/r (clipped), st = sqrt(1-ct^2)
    //   ca = cos(azim) = x/rxy, sa = sin(azim) = y/rxy
    float ct[2], st[2], ca[2], sa[2];
    #pragma unroll
    for (int v = 0; v < 2; ++v) {
        float x = xx[v], y = yy[v], z = zz[v];
        float rxy2 = fmaf(x, x, y*y);
        float r2   = fmaf(z, z, rxy2);
        float rinv = rsqrtf(fmaxf(r2, 1e-30f));
        float c    = fminf(1.f, fmaxf(-1.f, z * rinv));
        ct[v] = c;
        st[v] = sqrtf(fmaxf(1.f - c*c, 0.f));
        bool  zr   = (rxy2 <= 1e-30f);
        float ri2  = rsqrtf(fmaxf(rxy2, 1e-30f));
        ca[v] = zr ? 1.f : x * ri2;
        sa[v] = zr ? 0.f : y * ri2;
    }

    float* base = smem + tid*SLICE;

    // ---- Chebyshev recurrences for cos(m a), sin(m a) ---------------------
    #pragma unroll
    for (int v = 0; v < 2; ++v) {
        float* cs = base + v*VOFF + NP;
        float c1 = ca[v], s1 = sa[v];
        cs[0] = 1.f; cs[1] = c1; cs[21] = s1;
        float t2 = 2.f*c1;
        float cm2 = 1.f, cm1 = c1, sm2 = 0.f, sm1 = s1;
        #pragma unroll
        for (int m = 2; m <= LMAX; ++m) {
            float cm = fmaf(t2, cm1, -cm2);
            float sm = fmaf(t2, sm1, -sm2);
            cs[m] = cm; cs[20+m] = sm;
            cm2 = cm1; cm1 = cm; sm2 = sm1; sm1 = sm;
        }
    }

    // ---- associated Legendre, row-wise (l outer), both vectors interleaved
    float Pa[2][LMAX+1], Pb[2][LMAX+1];      // rows l-1 and l-2, in registers
    #pragma unroll
    for (int v = 0; v < 2; ++v) {
        base[v*VOFF + 0]  = 1.f;             // P(0,0)
        float p10 = ct[v];                   // P(1,0) = x
        float p11 = -st[v];                  // P(1,1) = -s  (Condon-Shortley)
        base[v*VOFF + 1]  = p10;             // pidx(1,0)
        base[v*VOFF + 21] = p11;             // pidx(1,1) = col_off(1)
        Pb[v][0] = 1.f; Pa[v][0] = p10; Pa[v][1] = p11;
    }
    #pragma unroll
    for (int l = 2; l <= LMAX; ++l) {
        #pragma unroll
        for (int v = 0; v < 2; ++v) {
            const float ctl = (float)(2*l-1) * ct[v];
            #pragma unroll
            for (int m = 0; m <= l-2; ++m) {
                float t  = ctl * Pa[v][m];
                float pn = fmaf(-(float)(l+m-1), Pb[v][m], t)
                         * (1.f/(float)(l-m));
                base[v*VOFF + col_off(m) + (l-m)] = pn;
                Pb[v][m] = Pa[v][m]; Pa[v][m] = pn;
            }
            float pd  = Pa[v][l-1];                     // P(l-1,l-1)
            float pl1 = ctl * pd;                       // P(l,l-1)
            float pll = -(float)(2*l-1) * st[v] * pd;   // P(l,l)
            base[v*VOFF + col_off(l-1) + 1] = pl1;
            base[v*VOFF + col_off(l)]       = pll;
            Pb[v][l-1] = pd; Pa[v][l-1] = pl1; Pa[v][l] = pll;
        }
    }

    // ---- rank gather: out[r] = C[r] * P1[i]*T1[i] * P2[j]*T2[j] -----------
    // Same-thread LDS RAW only (own slice) -> no barrier needed.
    if (active) {
        float4* op = (float4*)(out + n*(size_t)RANKN);
        #pragma unroll 4
        for (int r4 = 0; r4 < RANKN/4; ++r4) {
            int4 q0 = midx[4*r4+0], q1 = midx[4*r4+1];
            int4 q2 = midx[4*r4+2], q3 = midx[4*r4+3];
            float c0 = mcoef[4*r4+0], c1 = mcoef[4*r4+1];
            float c2 = mcoef[4*r4+2], c3 = mcoef[4*r4+3];
            float4 y;
            y.x = c0 * base[q0.x] * base[q0.y] * base[q0.z] * base[q0.w];
            y.y = c1 * base[q1.x] * base[q1.y] * base[q1.z] * base[q1.w];
            y.z = c2 * base[q2.x] * base[q2.y] * base[q2.z] * base[q2.w];
            y.w = c3 * base[q3.x] * base[q3.y] * base[q3.z] * base[q3.w];
            op[r4] = y;
        }
    }
}

// ------------------------------- launcher ----------------------------------
extern "C" void kernel_launch(void* const* d_in, const int* in_sizes, int n_in,
                              void* d_out, int out_size, void* d_ws, size_t ws_size,
                              hipStream_t stream)
{
    const float* coords = (const float*)d_in[0];   // [N,6] fp32
    const int*   ri     = (const int*)d_in[1];     // [256] i32
    const int*   rj     = (const int*)d_in[2];     // [256] i32
    float*       out    = (float*)d_out;           // [N,256] fp32

    int npts = in_sizes[0] / 6;

    int4*  midx  = (int4*)d_ws;                                  // 4 KB
    float* mcoef = (float*)((char*)d_ws + RANKN*sizeof(int4));   // 1 KB

    sh_meta<<<1, RANKN, 0, stream>>>(ri, rj, midx, mcoef);

    int grid = (npts + TPB - 1) / TPB;
    sh_main<<<grid, TPB, 0, stream>>>(coords, midx, mcoef, out, npts);
}